// AdaptiveHypergraphAttention_86861418594716
// MI455X (gfx1250) — compile-verified
//
#include <hip/hip_runtime.h>
#include <hip/hip_bf16.h>

// ---------------------------------------------------------------------------
// AdaptiveHypergraphAttention forward for MI455X (gfx1250, wave32, WMMA)
//
// Sizes (fixed by reference): S=716, M0=32 (M<=32 runtime), DM=64, B*L=192,
// nnz = S*K = 2148.  Dominant work = fp32 GEMM (192*716,64)x(64,64) done with
// V_WMMA_F32_16X16X4_F32, epilogue-fused transpose to xn[s][bl][d].
// ---------------------------------------------------------------------------

#define S_N   716
#define M0C   32
#define DMC   64
#define BLC   192          // B*L = 8*24
#define GAMMA 0.5f
#define SLOPE 0.02f

typedef float v2f __attribute__((ext_vector_type(2)));
typedef float v8f __attribute__((ext_vector_type(8)));

__device__ __forceinline__ float wave_sum(float v) {
#pragma unroll
    for (int off = 16; off > 0; off >>= 1) v += __shfl_xor(v, off, 32);
    return v;
}
__device__ __forceinline__ float wave_max(float v) {
#pragma unroll
    for (int off = 16; off > 0; off >>= 1) v = fmaxf(v, __shfl_xor(v, off, 32));
    return v;
}

// ---------------------------------------------------------------------------
// 0) Serial setup: segment offsets, degrees, node-CSR, zero loss accumulators.
//    nnz = 2148 -> microseconds, deterministic, graph-capture safe.
// ---------------------------------------------------------------------------
__global__ void setup_k(const int* __restrict__ node_idx,
                        const int* __restrict__ edge_idx,
                        int nnz,
                        int* __restrict__ cnt_e, float* __restrict__ b_inv,
                        int* __restrict__ edge_ptr,
                        int* __restrict__ deg,  float* __restrict__ dinv_sqrt,
                        int* __restrict__ node_ptr, int* __restrict__ node_csr,
                        float* __restrict__ loss_acc)
{
    if (threadIdx.x != 0 || blockIdx.x != 0) return;
    loss_acc[0] = 0.0f;            // sum |v - e_feat|
    loss_acc[1] = 0.0f;            // hyper loss
    for (int e = 0; e < M0C; ++e) cnt_e[e] = 0;
    for (int s = 0; s < S_N; ++s) deg[s] = 0;
    for (int i = 0; i < nnz; ++i) { cnt_e[edge_idx[i]]++; deg[node_idx[i]]++; }
    int acc = 0;
    for (int e = 0; e <= M0C; ++e) { edge_ptr[e] = acc; if (e < M0C) acc += cnt_e[e]; }
    for (int e = 0; e < M0C; ++e) b_inv[e] = cnt_e[e] > 0 ? 1.0f / (float)cnt_e[e] : 0.0f;
    acc = 0;
    for (int s = 0; s <= S_N; ++s) { node_ptr[s] = acc; if (s < S_N) acc += deg[s]; }
    for (int s = 0; s < S_N; ++s) dinv_sqrt[s] = deg[s] > 0 ? rsqrtf((float)deg[s]) : 0.0f;
    for (int s = 0; s < S_N; ++s) deg[s] = 0;              // reuse as cursor
    for (int i = 0; i < nnz; ++i) {
        int s = node_idx[i];
        node_csr[node_ptr[s] + deg[s]] = i;
        deg[s]++;
    }
    for (int s = 0; s < S_N; ++s) deg[s] = node_ptr[s + 1] - node_ptr[s]; // restore
}

// ---------------------------------------------------------------------------
// 1) GEMM + fused transpose:  xn[s][bl][d] = sum_k x[bl][s][k]*W[k][d] + bias[d]
//    One wave -> 16(bl) x 16(d) tile for node s, 16 chained f32 WMMAs (K=64).
//    grid.x = S * (BL/16), block = 128 (4 waves, wave w -> d-tile w).
// ---------------------------------------------------------------------------
__global__ void gemm_xw_wmma_k(const float* __restrict__ x,     // (BL,S,64)
                               const float* __restrict__ W,     // (64,64)
                               const float* __restrict__ bias,  // (64)
                               float* __restrict__ xn)          // (S,BL,64)
{
    const int tile   = blockIdx.x;
    const int s      = tile / (BLC / 16);
    const int blt    = tile % (BLC / 16);
    const int lane   = threadIdx.x & 31;
    const int wave   = threadIdx.x >> 5;
    const int dbase  = wave * 16;
    const int blbase = blt * 16;
    const int half   = lane >> 4;          // 0: K even-pair low, 1: high
    const int l15    = lane & 15;

    const float* __restrict__ xrow = x + ((size_t)(blbase + l15) * S_N + s) * DMC;

    v8f acc = {};
#pragma unroll
    for (int k0 = 0; k0 < DMC; k0 += 4) {
        const int ka = k0 + 2 * half;
        v2f a, b;
        a.x = xrow[ka];
        a.y = xrow[ka + 1];
        b.x = W[(size_t)ka       * DMC + dbase + l15];
        b.y = W[(size_t)(ka + 1) * DMC + dbase + l15];
        acc = __builtin_amdgcn_wmma_f32_16x16x4_f32(
            /*neg_a=*/false, a, /*neg_b=*/false, b,
            /*c_mod=*/(short)0, acc, /*reuse_a=*/false, /*reuse_b=*/false);
    }

    const float bv = bias[dbase + l15];
    float* __restrict__ outb = xn + ((size_t)s * BLC + blbase) * DMC + dbase;
#pragma unroll
    for (int i = 0; i < 8; ++i) {
        const int row = i + 8 * half;      // D layout: vgpr i, lanes>=16 -> M=8+i
        outb[(size_t)row * DMC + l15] = acc[i] + bv;
    }
}

// ---------------------------------------------------------------------------
// 2) res[e][b][d] = b_inv[e] * sum_{t in edge e} xn[node(t)][b][d]
//    grid (M0, BL/4), block 256 = 4 b x 64 d. Edge segments contiguous.
// ---------------------------------------------------------------------------
__global__ void res_k(const float* __restrict__ xn,
                      const int* __restrict__ node_idx,
                      const int* __restrict__ edge_ptr,
                      const float* __restrict__ b_inv,
                      const int* __restrict__ m_ptr,
                      float* __restrict__ res)
{
    const int e = blockIdx.x;
    if (e >= *m_ptr) return;
    const int b = blockIdx.y * 4 + (threadIdx.x >> 6);
    const int d = threadIdx.x & 63;
    const int beg = edge_ptr[e], end = edge_ptr[e + 1];
    float sum = 0.0f;
    for (int t = beg; t < end; ++t) {
        const int node = node_idx[t];
        sum += xn[((size_t)node * BLC + b) * DMC + d];
    }
    res[((size_t)e * BLC + b) * DMC + d] = sum * b_inv[e];
}

// ---------------------------------------------------------------------------
// 3) H_n logits + node loss.  One wave per (t,b); block 256 = 8 waves.
// ---------------------------------------------------------------------------
__global__ void hn_logits_k(const float* __restrict__ xn,
                            const float* __restrict__ res,
                            const int* __restrict__ node_idx,
                            const int* __restrict__ edge_idx,
                            const float* __restrict__ a_vec,   // 128
                            int nnz,
                            float* __restrict__ H_n,           // (nnz,BL)
                            float* __restrict__ loss_acc)
{
    __shared__ float blk;
    if (threadIdx.x == 0) blk = 0.0f;
    __syncthreads();

    const int wid  = blockIdx.x * 8 + (threadIdx.x >> 5);
    const int lane = threadIdx.x & 31;
    const int total = nnz * BLC;
    float labs_lane0 = 0.0f;
    bool valid = (wid < total);
    if (valid) {
        const int t = wid / BLC, b = wid % BLC;
        const int node = node_idx[t], e = edge_idx[t];
        const float* vr = xn  + ((size_t)node * BLC + b) * DMC;
        const float* er = res + ((size_t)e    * BLC + b) * DMC;
        const float v0 = vr[lane], v1 = vr[lane + 32];
        const float e0 = er[lane], e1 = er[lane + 32];
        float labs = fabsf(v0 - e0) + fabsf(v1 - e1);
        float dot  = v0 * a_vec[lane]      + v1 * a_vec[lane + 32]
                   + e0 * a_vec[64 + lane] + e1 * a_vec[96 + lane];
        labs = wave_sum(labs);
        dot  = wave_sum(dot);
        if (lane == 0) {
            H_n[(size_t)t * BLC + b] = dot > 0.0f ? dot : SLOPE * dot;
            labs_lane0 = labs;
        }
    }
    if (valid && lane == 0) atomicAdd(&blk, labs_lane0);
    __syncthreads();
    if (threadIdx.x == 0) atomicAdd(&loss_acc[0], blk);
}

// ---------------------------------------------------------------------------
// 4) Segment softmax of H_n over contiguous edge segments, per b.
//    grid (M0, BL/8), block 256 = 8 waves, wave -> one b.
// ---------------------------------------------------------------------------
__global__ void hn_softmax_k(const int* __restrict__ edge_ptr,
                             const int* __restrict__ m_ptr,
                             float* __restrict__ H_n)
{
    const int e = blockIdx.x;
    if (e >= *m_ptr) return;
    const int b    = blockIdx.y * 8 + (threadIdx.x >> 5);
    const int lane = threadIdx.x & 31;
    const int beg = edge_ptr[e], end = edge_ptr[e + 1];

    float mx = -3.4e38f;
    for (int t = beg + lane; t < end; t += 32) mx = fmaxf(mx, H_n[(size_t)t * BLC + b]);
    mx = wave_max(mx);
    if (!(mx > -3.4e38f)) mx = 0.0f;   // empty-segment guard (isfinite in ref)

    float sum = 0.0f;
    for (int t = beg + lane; t < end; t += 32) {
        const float ex = expf(H_n[(size_t)t * BLC + b] - mx);
        H_n[(size_t)t * BLC + b] = ex;
        sum += ex;
    }
    sum = wave_sum(sum);
    const float inv = 1.0f / (sum + 1e-16f);
    for (int t = beg + lane; t < end; t += 32) H_n[(size_t)t * BLC + b] *= inv;
}

// ---------------------------------------------------------------------------
// 5) out1[e][b][d] = sum_t sqrt(d_inv)[node(t)] * xn[node(t)][b][d] * Hn[t][b]
// ---------------------------------------------------------------------------
__global__ void out1_k(const float* __restrict__ xn,
                       const float* __restrict__ H_n,
                       const int* __restrict__ node_idx,
                       const int* __restrict__ edge_ptr,
                       const float* __restrict__ dinv_sqrt,
                       const int* __restrict__ m_ptr,
                       float* __restrict__ out1)
{
    const int e = blockIdx.x;
    if (e >= *m_ptr) return;
    const int b = blockIdx.y * 4 + (threadIdx.x >> 6);
    const int d = threadIdx.x & 63;
    const int beg = edge_ptr[e], end = edge_ptr[e + 1];
    float sum = 0.0f;
    for (int t = beg; t < end; ++t) {
        const int node = node_idx[t];
        const float w = dinv_sqrt[node] * H_n[(size_t)t * BLC + b];
        sum += w * xn[((size_t)node * BLC + b) * DMC + d];
    }
    out1[((size_t)e * BLC + b) * DMC + d] = sum;
}

// ---------------------------------------------------------------------------
// 6a) nrm[e][b] = ||res[e][b][:]||   (wave per (e,b))
// ---------------------------------------------------------------------------
__global__ void nrm_k(const float* __restrict__ res,
                      const int* __restrict__ m_ptr,
                      float* __restrict__ nrm)
{
    const int wid  = blockIdx.x * 8 + (threadIdx.x >> 5);
    const int lane = threadIdx.x & 31;
    if (wid >= M0C * BLC) return;
    if (wid / BLC >= *m_ptr) return;
    const float* r = res + (size_t)wid * DMC;
    float s = r[lane] * r[lane] + r[lane + 32] * r[lane + 32];
    s = wave_sum(s);
    if (lane == 0) nrm[wid] = sqrtf(s);
}

// ---------------------------------------------------------------------------
// 6b) hyper loss: per (e,m) pair, mean_b[ alpha*dist + (1-alpha)*relu(G-dist) ]
//     grid (M0,M0), block 256 = 8 waves over b.
// ---------------------------------------------------------------------------
__global__ void hyper_loss_k(const float* __restrict__ res,
                             const float* __restrict__ nrm,
                             const int* __restrict__ m_ptr,
                             float* __restrict__ loss_acc)
{
    const int M = *m_ptr;
    const int e = blockIdx.x, m = blockIdx.y;
    if (e >= M || m >= M) return;
    const int wave = threadIdx.x >> 5, lane = threadIdx.x & 31;
    __shared__ float part[8];

    float acc = 0.0f;
    for (int b = wave; b < BLC; b += 8) {
        const float* re = res + ((size_t)e * BLC + b) * DMC;
        const float* rm = res + ((size_t)m * BLC + b) * DMC;
        const float x0 = re[lane], x1 = re[lane + 32];
        const float y0 = rm[lane], y1 = rm[lane + 32];
        float dot = x0 * y0 + x1 * y1;
        const float d0 = x0 - y0, d1 = x1 - y1;
        float sq = d0 * d0 + d1 * d1;
        dot = wave_sum(dot);
        sq  = wave_sum(sq);
        if (lane == 0) {
            const float ne = fmaxf(nrm[e * BLC + b], 1e-12f);
            const float nm = fmaxf(nrm[m * BLC + b], 1e-12f);
            const float alpha = dot / (ne * nm);
            const float dist  = sqrtf(sq + 1e-8f);
            const float hinge = fmaxf(GAMMA - dist, 0.0f);
            acc += alpha * dist + (1.0f - alpha) * hinge;
        }
    }
    if (lane == 0) part[wave] = acc;
    __syncthreads();
    if (threadIdx.x == 0) {
        float s = 0.0f;
        for (int i = 0; i < 8; ++i) s += part[i];
        const float P = s / (float)BLC;                 // mean over b
        atomicAdd(&loss_acc[1], fabsf(P) / ((float)M * (float)M));
    }
}

// ---------------------------------------------------------------------------
// 7) H_e logits: leaky( dot(out1[e(t)][b], bvec[0:64]) + dot(v, bvec[64:128]) )
// ---------------------------------------------------------------------------
__global__ void he_logits_k(const float* __restrict__ xn,
                            const float* __restrict__ out1,
                            const int* __restrict__ node_idx,
                            const int* __restrict__ edge_idx,
                            const float* __restrict__ b_vec,   // 128
                            int nnz,
                            float* __restrict__ H_e)           // (nnz,BL)
{
    const int wid  = blockIdx.x * 8 + (threadIdx.x >> 5);
    const int lane = threadIdx.x & 31;
    if (wid >= nnz * BLC) return;
    const int t = wid / BLC, b = wid % BLC;
    const int node = node_idx[t], e = edge_idx[t];
    const float* e2 = out1 + ((size_t)e    * BLC + b) * DMC;
    const float* vr = xn   + ((size_t)node * BLC + b) * DMC;
    float dot = e2[lane] * b_vec[lane]      + e2[lane + 32] * b_vec[lane + 32]
              + vr[lane] * b_vec[64 + lane] + vr[lane + 32] * b_vec[96 + lane];
    dot = wave_sum(dot);
    if (lane == 0) H_e[(size_t)t * BLC + b] = dot > 0.0f ? dot : SLOPE * dot;
}

// ---------------------------------------------------------------------------
// 8) Segment softmax of H_e over node segments (K<=3 entries via node CSR).
//    Thread per (node,b): coalesced in b.
// ---------------------------------------------------------------------------
__global__ void he_softmax_k(const int* __restrict__ node_ptr,
                             const int* __restrict__ node_csr,
                             float* __restrict__ H_e)
{
    const int idx = blockIdx.x * 256 + threadIdx.x;
    if (idx >= S_N * BLC) return;
    const int s = idx / BLC, b = idx % BLC;
    const int beg = node_ptr[s], end = node_ptr[s + 1];
    float mx = -3.4e38f;
    for (int j = beg; j < end; ++j)
        mx = fmaxf(mx, H_e[(size_t)node_csr[j] * BLC + b]);
    if (!(mx > -3.4e38f)) mx = 0.0f;
    float sum = 0.0f;
    for (int j = beg; j < end; ++j) {
        const size_t p = (size_t)node_csr[j] * BLC + b;
        const float ex = expf(H_e[p] - mx);
        H_e[p] = ex;
        sum += ex;
    }
    const float inv = 1.0f / (sum + 1e-16f);
    for (int j = beg; j < end; ++j) H_e[(size_t)node_csr[j] * BLC + b] *= inv;
}

// ---------------------------------------------------------------------------
// 9) Final: out[bl][s][d] = sqrt(d_inv)[s] *
//        sum_{t in node s} b_inv[e(t)] * out1[e(t)][b][d] * He[t][b]
//    Block per node; 256 threads over (b,d).
// ---------------------------------------------------------------------------
__global__ void out_final_k(const float* __restrict__ out1,
                            const float* __restrict__ H_e,
                            const int* __restrict__ node_ptr,
                            const int* __restrict__ node_csr,
                            const int* __restrict__ edge_idx,
                            const float* __restrict__ b_inv,
                            const float* __restrict__ dinv_sqrt,
                            float* __restrict__ out)
{
    const int s = blockIdx.x;
    const int beg = node_ptr[s], end = node_ptr[s + 1];
    const float ds = dinv_sqrt[s];
    for (int idx = threadIdx.x; idx < BLC * DMC; idx += 256) {
        const int b = idx >> 6, d = idx & 63;
        float sum = 0.0f;
        for (int j = beg; j < end; ++j) {
            const int t = node_csr[j];
            const int e = edge_idx[t];
            sum += b_inv[e] * H_e[(size_t)t * BLC + b]
                 * out1[((size_t)e * BLC + b) * DMC + d];
        }
        out[((size_t)b * S_N + s) * DMC + d] = sum * ds;
    }
}

// ---------------------------------------------------------------------------
// 10) loss = loss_node/(nnz*BL*64) + loss_hyper
// ---------------------------------------------------------------------------
__global__ void finalize_k(const float* __restrict__ loss_acc, int nnz,
                           float* __restrict__ loss_out)
{
    if (threadIdx.x == 0 && blockIdx.x == 0)
        *loss_out = loss_acc[0] / ((float)nnz * (float)BLC * (float)DMC)
                  + loss_acc[1];
}

// ---------------------------------------------------------------------------
extern "C" void kernel_launch(void* const* d_in, const int* in_sizes, int n_in,
                              void* d_out, int out_size, void* d_ws, size_t ws_size,
                              hipStream_t stream)
{
    const float* x        = (const float*)d_in[0];   // (B,L,S,64)
    const int*   node_idx = (const int*)  d_in[1];   // (nnz)
    const int*   edge_idx = (const int*)  d_in[2];   // (nnz)
    const int*   m_ptr    = (const int*)  d_in[3];   // scalar M
    const float* weight   = (const float*)d_in[4];   // (64,64)
    const float* bias     = (const float*)d_in[5];   // (64)
    const float* a_vec    = (const float*)d_in[6];   // (128,1)
    const float* b_vec    = (const float*)d_in[7];   // (128,1)
    float*       out      = (float*)d_out;
    const int    nnz      = in_sizes[1];

    // Workspace carve (256B aligned). Total ~42 MB.
    char* w = (char*)d_ws;
    auto carve = [&](size_t bytes) {
        void* p = (void*)w;
        w += (bytes + 255) & ~(size_t)255;
        return p;
    };
    float* xn        = (float*)carve(sizeof(float) * (size_t)S_N * BLC * DMC);
    float* res       = (float*)carve(sizeof(float) * (size_t)M0C * BLC * DMC);
    float* out1      = (float*)carve(sizeof(float) * (size_t)M0C * BLC * DMC);
    float* H_n       = (float*)carve(sizeof(float) * (size_t)nnz * BLC);
    float* H_e       = (float*)carve(sizeof(float) * (size_t)nnz * BLC);
    float* nrm       = (float*)carve(sizeof(float) * (size_t)M0C * BLC);
    int*   cnt_e     = (int*)  carve(sizeof(int)   * M0C);
    float* b_inv     = (float*)carve(sizeof(float) * M0C);
    int*   edge_ptr  = (int*)  carve(sizeof(int)   * (M0C + 1));
    int*   deg       = (int*)  carve(sizeof(int)   * S_N);
    float* dinv_sqrt = (float*)carve(sizeof(float) * S_N);
    int*   node_ptr  = (int*)  carve(sizeof(int)   * (S_N + 1));
    int*   node_csr  = (int*)  carve(sizeof(int)   * nnz);
    float* loss_acc  = (float*)carve(sizeof(float) * 2);

    float* loss_out = out + (out_size - 1);   // tuple: [out tensor ..., loss]

    // 0) serial setup
    setup_k<<<1, 1, 0, stream>>>(node_idx, edge_idx, nnz, cnt_e, b_inv, edge_ptr,
                                 deg, dinv_sqrt, node_ptr, node_csr, loss_acc);

    // 1) WMMA GEMM + transpose
    gemm_xw_wmma_k<<<S_N * (BLC / 16), 128, 0, stream>>>(x, weight, bias, xn);

    // 2) res
    res_k<<<dim3(M0C, BLC / 4), 256, 0, stream>>>(xn, node_idx, edge_ptr, b_inv,
                                                  m_ptr, res);

    // 3) H_n logits + node loss
    {
        const int waves = nnz * BLC;
        hn_logits_k<<<(waves + 7) / 8, 256, 0, stream>>>(xn, res, node_idx, edge_idx,
                                                         a_vec, nnz, H_n, loss_acc);
    }

    // 4) edge softmax
    hn_softmax_k<<<dim3(M0C, BLC / 8), 256, 0, stream>>>(edge_ptr, m_ptr, H_n);

    // 5) out1
    out1_k<<<dim3(M0C, BLC / 4), 256, 0, stream>>>(xn, H_n, node_idx, edge_ptr,
                                                   dinv_sqrt, m_ptr, out1);

    // 6) hyper loss
    nrm_k<<<(M0C * BLC + 7) / 8, 256, 0, stream>>>(res, m_ptr, nrm);
    hyper_loss_k<<<dim3(M0C, M0C), 256, 0, stream>>>(res, nrm, m_ptr, loss_acc);

    // 7) H_e logits
    {
        const int waves = nnz * BLC;
        he_logits_k<<<(waves + 7) / 8, 256, 0, stream>>>(xn, out1, node_idx, edge_idx,
                                                         b_vec, nnz, H_e);
    }

    // 8) node softmax
    he_softmax_k<<<(S_N * BLC + 255) / 256, 256, 0, stream>>>(node_ptr, node_csr, H_e);

    // 9) final output
    out_final_k<<<S_N, 256, 0, stream>>>(out1, H_e, node_ptr, node_csr, edge_idx,
                                         b_inv, dinv_sqrt, out);

    // 10) scalar loss
    finalize_k<<<1, 1, 0, stream>>>(loss_acc, nnz, loss_out);
}